// CenterNetLoss_11046655885666
// MI455X (gfx1250) — compile-verified
//
#include <hip/hip_runtime.h>
#include <hip/hip_bf16.h>
#include <stdint.h>

#define NUM_CLASSES 80
#define FMAP_H 128
#define FMAP_W 128
#define BATCH 16
#define MAX_OBJ 100
#define HW (FMAP_H * FMAP_W)
#define TILE 4096
#define MAX_LIST 32
#define FACTOR (1.0f / 9.0f)

typedef __attribute__((ext_vector_type(2))) float v2f;
typedef __attribute__((ext_vector_type(8))) float v8f;
typedef __attribute__((ext_vector_type(4))) unsigned int v4u;
typedef __attribute__((ext_vector_type(8))) unsigned int v8u;

// ---------------- init: zero accumulators + per-(b,c) counts ----------------
__global__ void cnl_init(float* acc, int* counts) {
    int t = threadIdx.x;
    if (t < 8) acc[t] = 0.0f;
    for (int i = t; i < BATCH * NUM_CLASSES; i += blockDim.x) counts[i] = 0;
}

// ---------------- per-object target prep + per-class compaction -------------
__global__ void cnl_prep(const float* __restrict__ ann, float4* objHM, float4* objREG,
                         int* objIDX, int* counts, int* list) {
    int b = blockIdx.x, k = threadIdx.x;
    if (k >= MAX_OBJ) return;
    const float* a = ann + (b * MAX_OBJ + k) * 5;
    float clsf = a[4];
    bool valid = clsf >= 0.0f;
    float x1 = fminf(fmaxf(a[0] * 0.25f, 0.0f), FMAP_W - 1.0f);
    float y1 = fminf(fmaxf(a[1] * 0.25f, 0.0f), FMAP_H - 1.0f);
    float x2 = fminf(fmaxf(a[2] * 0.25f, 0.0f), FMAP_W - 1.0f);
    float y2 = fminf(fmaxf(a[3] * 0.25f, 0.0f), FMAP_H - 1.0f);
    float bw = x2 - x1, bh = y2 - y1;
    float cx = 0.5f * (x1 + x2), cy = 0.5f * (y1 + y2);
    float cxi = truncf(cx), cyi = truncf(cy);
    // gaussian radius (MIN_OVERLAP = 0.7)
    float h = ceilf(bh), w = ceilf(bw);
    float b1 = h + w, c1 = w * h * (0.3f / 1.7f);
    float r1 = 0.5f * (b1 + sqrtf(fmaxf(b1 * b1 - 4.0f * c1, 0.0f)));
    float b2 = 2.0f * (h + w), c2 = 0.3f * w * h;
    float r2 = 0.5f * (b2 + sqrtf(fmaxf(b2 * b2 - 16.0f * c2, 0.0f)));
    float a3 = 2.8f, b3 = -1.4f * (h + w), c3 = -0.3f * w * h;
    float r3 = (b3 + sqrtf(fmaxf(b3 * b3 - 4.0f * a3 * c3, 0.0f))) / (2.0f * a3);
    float r = fmaxf(0.0f, truncf(fminf(fminf(r1, r2), r3)));
    float sigma = (2.0f * r + 1.0f) / 6.0f;
    float inv2s2 = 1.0f / (2.0f * sigma * sigma);
    int gi = b * MAX_OBJ + k;
    objHM[gi] = make_float4(cxi, cyi, r, inv2s2);
    objREG[gi] = make_float4(cx - cxi, cy - cyi, bw, bh);
    int ridx = (int)(cyi * (float)FMAP_W + cxi);
    objIDX[gi] = valid ? ridx : -1;
    if (valid) {
        int c = (int)clsf;
        int slot = atomicAdd(&counts[b * NUM_CLASSES + c], 1);
        if (slot < MAX_LIST) list[(b * NUM_CLASSES + c) * MAX_LIST + slot] = k;
    }
}

// ---------------- focal (heatmap) loss: TDM-staged tiles + WMMA reduce ------
__global__ __launch_bounds__(256) void cnl_hm(const float* __restrict__ hm,
                                              const float4* __restrict__ objHM,
                                              const int* __restrict__ counts,
                                              const int* __restrict__ list,
                                              float* acc) {
    __shared__ __align__(16) float tile[TILE];   // 16 KB staged via TDM
    __shared__ float4 objs[MAX_LIST];
    __shared__ float redL[16];
    __shared__ float redC[16];

    int b = blockIdx.z, c = blockIdx.y;
    int tileStart = blockIdx.x * TILE;
    int bc = b * NUM_CLASSES + c;
    int n = counts[bc];
    n = n < MAX_LIST ? n : MAX_LIST;
    if ((int)threadIdx.x < n) {
        int k = list[bc * MAX_LIST + threadIdx.x];
        objs[threadIdx.x] = objHM[b * MAX_OBJ + k];
    }

    // Tensor Data Mover: 1D tile, 4096 f32 elements, global -> LDS.
    // Issued by wave 0 only (EXEC ignored for tensor ops -> one DMA per wave).
    if (threadIdx.x < 32) {
        const float* src = hm + ((size_t)bc * HW + tileStart);
        uint64_t ga = (uint64_t)(uintptr_t)src;
        uint32_t lds = (uint32_t)(uintptr_t)&tile[0];  // low 32 bits = LDS offset
        v4u g0;
        g0[0] = 1u;                                    // count=1 valid descriptor
        g0[1] = lds;                                   // lds_addr
        g0[2] = (uint32_t)ga;                          // global_addr[31:0]
        g0[3] = (uint32_t)(ga >> 32) | 0x80000000u;    // addr[56:32] | type=2
        v8u g1;
        g1[0] = 0x00020000u;                 // data_size=2 (4 bytes)
        g1[1] = (uint32_t)TILE << 16;        // tensor_dim0 = 4096
        g1[2] = 1u << 16;                    // tensor_dim1 = 1
        g1[3] = (uint32_t)TILE << 16;        // tile_dim0 = 4096 (1D tile)
        g1[4] = 0u;                          // tile_dim1/2 unused
        g1[5] = (uint32_t)TILE;              // tensor_dim0_stride
        g1[6] = 0u;
        g1[7] = 0u;
        asm volatile("tensor_load_to_lds %0, %1" :: "s"(g0), "s"(g1) : "memory");
        __builtin_amdgcn_s_wait_tensorcnt(0);
    }
    __syncthreads();

    float loss = 0.0f, cnt = 0.0f;
    const float4* t4 = (const float4*)tile;
#pragma unroll
    for (int i = 0; i < 4; ++i) {
        int vidx = threadIdx.x + i * 256;     // float4 index -> ds_load_b128
        float4 v4 = t4[vidx];
        int pbase = tileStart + vidx * 4;
        float vals[4] = {v4.x, v4.y, v4.z, v4.w};
#pragma unroll
        for (int e = 0; e < 4; ++e) {
            int p = pbase + e;
            float xf = (float)(p & (FMAP_W - 1));
            float yf = (float)(p >> 7);
            float hmt = 0.0f;
            for (int j = 0; j < n; ++j) {     // avg ~1.25 objects per class
                float4 o = objs[j];
                float dx = xf - o.x, dy = yf - o.y;
                bool inb = (fabsf(dx) <= o.z) & (fabsf(dy) <= o.z);
                float g = __expf(-(dx * dx + dy * dy) * o.w);
                hmt = inb ? fmaxf(hmt, g) : hmt;
            }
            float pred = fminf(fmaxf(vals[e], 1e-4f), 1.0f - 1e-4f);
            if (hmt == 1.0f) {
                float om = 1.0f - pred;
                loss += -__logf(pred) * om * om;
                cnt += 1.0f;
            } else {
                float q = 1.0f - hmt;
                float q2 = q * q;
                loss += -__logf(1.0f - pred) * pred * pred * (q2 * q2);
            }
        }
    }

    // Single V_WMMA_F32_16X16X4_F32 reduces BOTH accumulators per wave:
    //   A[M][0]=loss(lane M), A[M][1]=cnt(lane M),
    //   A[M][2]=loss(lane M+16), A[M][3]=cnt(lane M+16)
    //   B column-selector: cols 0-7 pick K=0,2 (loss); cols 8-15 pick K=1,3 (cnt)
    // => D[M][j<8] = loss(M)+loss(M+16), D[M][j>=8] = cnt(M)+cnt(M+16).
    // After horizontal add of the 8 D VGPRs:
    //   lane 0 / 16 hold the two loss half-sums, lane 8 / 24 the count half-sums.
    int lane = threadIdx.x & 31, wave = threadIdx.x >> 5;
    float sel = ((lane & 15) < 8) ? 1.0f : 0.0f;
    v2f a;  a.x = loss;  a.y = cnt;
    v2f bs; bs.x = sel;  bs.y = 1.0f - sel;
    v8f cm = {};
    cm = __builtin_amdgcn_wmma_f32_16x16x4_f32(false, a, false, bs,
                                               (short)0, cm, false, false);
    float s = cm[0] + cm[1] + cm[2] + cm[3] + cm[4] + cm[5] + cm[6] + cm[7];
    if ((lane & 7) == 0) {                    // lanes 0, 8, 16, 24
        int half = lane >> 4;
        if (lane & 8) redC[wave * 2 + half] = s;
        else          redL[wave * 2 + half] = s;
    }
    __syncthreads();
    if (threadIdx.x == 0) {
        float L = 0.0f, C = 0.0f;
        for (int i = 0; i < 16; ++i) { L += redL[i]; C += redC[i]; }
        atomicAdd(&acc[0], L);
        atomicAdd(&acc[1], C);
    }
}

// ---------------- smooth-L1 on gathered offsets / wh -----------------------
__global__ void cnl_reg(const float* __restrict__ off, const float* __restrict__ wh,
                        const float4* __restrict__ objREG, const int* __restrict__ objIDX,
                        float* acc) {
    int b = blockIdx.x, k = threadIdx.x;
    if (k >= MAX_OBJ) return;
    int gi = b * MAX_OBJ + k;
    int idx = objIDX[gi];
    if (idx < 0) return;
    float4 t = objREG[gi];
    const float* ob = off + (size_t)b * 2 * HW;
    const float* wb = wh + (size_t)b * 2 * HW;
    float g0 = ob[idx], g1 = ob[HW + idx];
    float w0 = wb[idx], w1 = wb[HW + idx];
    auto sl1 = [](float d) {
        float x = fabsf(d);
        return x >= FACTOR ? x - 0.5f * FACTOR : 0.5f * x * x * (1.0f / FACTOR);
    };
    atomicAdd(&acc[2], sl1(g0 - t.x) + sl1(g1 - t.y));
    atomicAdd(&acc[3], sl1(w0 - t.z) + sl1(w1 - t.w));
    atomicAdd(&acc[4], 1.0f);
}

// ---------------- finalize 3 weighted losses --------------------------------
__global__ void cnl_final(const float* acc, float* out) {
    float posn = acc[1];
    float hm = posn > 0.0f ? acc[0] / fmaxf(posn, 1.0f) : 0.0f;
    float nv = acc[4];
    float offl = nv > 0.0f ? acc[2] / fmaxf(nv, 1.0f) : 0.0f;
    float whl  = nv > 0.0f ? acc[3] / fmaxf(nv, 1.0f) : 0.0f;
    out[0] = 1.0f * hm;    // HM_W
    out[1] = 1.0f * offl;  // OFF_W
    out[2] = 0.1f * whl;   // WH_W
}

extern "C" void kernel_launch(void* const* d_in, const int* in_sizes, int n_in,
                              void* d_out, int out_size, void* d_ws, size_t ws_size,
                              hipStream_t stream) {
    const float* hm  = (const float*)d_in[0];
    const float* off = (const float*)d_in[1];
    const float* wh  = (const float*)d_in[2];
    const float* ann = (const float*)d_in[3];
    float* out = (float*)d_out;

    // workspace layout (~222 KB total)
    char* ws = (char*)d_ws;
    float* acc  = (float*)ws;                                   // 8 floats
    int* counts = (int*)(ws + 256);                             // 1280 ints
    int* list   = (int*)(ws + 256 + 1280 * 4);                  // 16*80*32 ints
    char* p2 = ws + 256 + 1280 * 4 + BATCH * NUM_CLASSES * MAX_LIST * 4;
    float4* objHM  = (float4*)p2;                               // 1600 float4
    float4* objREG = objHM + BATCH * MAX_OBJ;                   // 1600 float4
    int* objIDX    = (int*)(objREG + BATCH * MAX_OBJ);          // 1600 ints

    cnl_init<<<1, 256, 0, stream>>>(acc, counts);
    cnl_prep<<<BATCH, 128, 0, stream>>>(ann, objHM, objREG, objIDX, counts, list);
    cnl_hm<<<dim3(HW / TILE, NUM_CLASSES, BATCH), 256, 0, stream>>>(hm, objHM, counts, list, acc);
    cnl_reg<<<BATCH, 128, 0, stream>>>(off, wh, objREG, objIDX, acc);
    cnl_final<<<1, 1, 0, stream>>>(acc, out);
}